// SSKernelNPLR_82394652607281
// MI455X (gfx1250) — compile-verified
//
#include <hip/hip_runtime.h>
#include <math.h>

// ---------------------------------------------------------------------------
// S4 NPLR kernel for MI455X (gfx1250, wave32).
//   Phase 1: Cauchy kernel + rank-1 Woodbury + bilinear factor -> Hermitian
//            spectrum S (scaled by 1/L), stored in LDS in four-step layout
//            S'[l1][l2] = S[l1 + 64*l2].
//   Phase 2: G[l1,t2] = sum_l2 S'[l1,l2] * e^{+2pi i l2 t2/128}   (WMMA f32)
//            then twiddle by e^{+2pi i l1 t2 / 8192}.
//   Phase 3: k[t2+128*t1] = Re sum_l1 e^{+2pi i t1 l1/64} G'[l1,t2] (WMMA f32)
// One workgroup per head; 256 threads = 8 wave32s; 128KB+ LDS (CDNA5 WGP has
// 320KB, which is what lets the whole per-head pipeline fuse into one kernel).
// ---------------------------------------------------------------------------

typedef __attribute__((ext_vector_type(2))) float v2f;
typedef __attribute__((ext_vector_type(8))) float v8f;

#define HH   256
#define NN   64
#define LLEN 8192
#define N2   128   // l2 / t2 extent
#define PI2F 6.28318530717958647692f

__device__ __forceinline__ v8f wmma4(v2f a, v2f b, v8f c) {
  // V_WMMA_F32_16X16X4_F32 : D(16x16 f32) = A(16x4) * B(4x16) + C
  return __builtin_amdgcn_wmma_f32_16x16x4_f32(
      false, a, false, b, (short)0, c, false, false);
}

__global__ __launch_bounds__(256) void s4_nplr_fused(
    const float* __restrict__ w_ri,  // (H,N,2)
    const float* __restrict__ p_ri,  // (H,1,N,2)
    const float* __restrict__ q_ri,  // (H,1,N,2)
    const float* __restrict__ B_ri,  // (H,N,2)
    const float* __restrict__ C_ri,  // (H,1,N,2)
    const float* __restrict__ log_dt,// (H,)
    float* __restrict__ out)         // (H, L)
{
  __shared__ float sSr[LLEN];   // spectrum real, layout [l1*128 + l2]
  __shared__ float sSi[LLEN];   // spectrum imag
  __shared__ float sGr[LLEN];   // G' real,     layout [l1*128 + t2]
  __shared__ float sGi[LLEN];   // G' imag
  __shared__ float sPar[5 * 2 * NN]; // w | B | C | p | q, interleaved re/im

  const int h    = blockIdx.x;
  const int tid  = threadIdx.x;
  const int lane = tid & 31;
  const int wave = tid >> 5;

  // ---- stage per-head parameters into LDS -------------------------------
  if (tid < NN) {
    const int n = tid;
    const int g = (h * NN + n) * 2;
    sPar[0 * 128 + 2 * n] = w_ri[g];     sPar[0 * 128 + 2 * n + 1] = w_ri[g + 1];
    sPar[1 * 128 + 2 * n] = B_ri[g];     sPar[1 * 128 + 2 * n + 1] = B_ri[g + 1];
    sPar[2 * 128 + 2 * n] = C_ri[g];     sPar[2 * 128 + 2 * n + 1] = C_ri[g + 1];
    sPar[3 * 128 + 2 * n] = p_ri[g];     sPar[3 * 128 + 2 * n + 1] = p_ri[g + 1];
    sPar[4 * 128 + 2 * n] = q_ri[g];     sPar[4 * 128 + 2 * n + 1] = q_ri[g + 1];
  }
  const float dt = expf(log_dt[h]);
  __syncthreads();

  // ---- Phase 1: Cauchy + Woodbury -> spectrum in LDS --------------------
  const float thscale = PI2F / (float)LLEN;
  for (int l = tid; l <= LLEN / 2; l += 256) {
    float sn, cs;
    __sincosf(thscale * (float)l, &sn, &cs);
    const float nr = cs, ni = -sn;            // node = exp(-2pi i l / L)
    const float dr = 1.0f + nr, di = ni;      // 1 + node
    const float invd2 = 1.0f / (dr * dr + di * di);
    // z = 2(1-node)/(1+node)
    const float numr = 2.0f * (1.0f - nr), numi = -2.0f * ni;
    const float zr = (numr * dr + numi * di) * invd2;
    const float zi = (numi * dr - numr * di) * invd2;
    // bilinear factor 2/(1+node)
    const float wfr = 2.0f * dr * invd2, wfi = -2.0f * di * invd2;

    float r00r = 0.f, r00i = 0.f, r01r = 0.f, r01i = 0.f;
    float r10r = 0.f, r10i = 0.f, r11r = 0.f, r11i = 0.f;
    for (int n = 0; n < NN; ++n) {
      const float wr = sPar[2 * n] * dt, wi = sPar[2 * n + 1] * dt;
      const float ax = zr - wr;
      const float b1 = zi - wi;
      const float i1 = 1.0f / (ax * ax + b1 * b1);
      const float cwr = ax * i1, cwi = -b1 * i1;      // 1/(z - w dt)
      const float b2 = zi + wi;
      const float i2 = 1.0f / (ax * ax + b2 * b2);
      const float ccr = ax * i2, cci = -b2 * i2;      // 1/(z - conj(w dt))
      const float sr_ = cwr + ccr, si_ = cwi + cci;   // s = cw + cwc
      const float dfr = cwr - ccr, dfi = cwi - cci;   // d = cw - cwc
      // v*cw + conj(v)*cwc = Re(v)*s + i*Im(v)*d
      const float Br = sPar[128 + 2 * n], Bi = sPar[128 + 2 * n + 1];
      const float Cr = sPar[256 + 2 * n], Ci = sPar[256 + 2 * n + 1];
      const float pr = sPar[384 + 2 * n], pi = sPar[384 + 2 * n + 1];
      const float qr = sPar[512 + 2 * n], qi = sPar[512 + 2 * n + 1];
      float vr, vi;
      vr = Br * Cr + Bi * Ci; vi = Bi * Cr - Br * Ci;   // B * conj(C)
      r00r += vr * sr_ - vi * dfi; r00i += vr * si_ + vi * dfr;
      vr = pr * Cr + pi * Ci; vi = pi * Cr - pr * Ci;   // p * conj(C)
      r01r += vr * sr_ - vi * dfi; r01i += vr * si_ + vi * dfr;
      vr = Br * qr + Bi * qi; vi = Bi * qr - Br * qi;   // B * conj(q)
      r10r += vr * sr_ - vi * dfi; r10i += vr * si_ + vi * dfr;
      vr = pr * qr + pi * qi; vi = pi * qr - pr * qi;   // p * conj(q)
      r11r += vr * sr_ - vi * dfi; r11i += vr * si_ + vi * dfr;
    }
    r00r *= dt; r00i *= dt; r01r *= dt; r01i *= dt;
    r10r *= dt; r10i *= dt; r11r *= dt; r11i *= dt;
    // k_f = r00 - r01*r10/(1+r11)
    const float e_r = 1.0f + r11r, e_i = r11i;
    const float ie2 = 1.0f / (e_r * e_r + e_i * e_i);
    const float mr = r01r * r10r - r01i * r10i;
    const float mi = r01r * r10i + r01i * r10r;
    float kfr = r00r - (mr * e_r + mi * e_i) * ie2;
    float kfi = r00i - (mi * e_r - mr * e_i) * ie2;
    // * 2/(1+node), * 1/L (iDFT normalization)
    const float sr2 = (kfr * wfr - kfi * wfi) * (1.0f / (float)LLEN);
    const float si2 = (kfr * wfi + kfi * wfr) * (1.0f / (float)LLEN);
    const int idx = (l & 63) * N2 + (l >> 6);
    sSr[idx] = sr2; sSi[idx] = si2;
    if (l > 0 && l < LLEN / 2) {            // Hermitian mirror (written once)
      const int lm = LLEN - l;
      const int idxm = (lm & 63) * N2 + (lm >> 6);
      sSr[idxm] = sr2; sSi[idxm] = -si2;
    }
  }
  __syncthreads();

  // Fragment geometry shared by A and B operands of V_WMMA_F32_16X16X4_F32:
  //   lanes 0-15  : row/col = lane,    holds K = k0, k0+1
  //   lanes 16-31 : row/col = lane-16, holds K = k0+2, k0+3
  const int frow = lane & 15;
  const int koff = (lane >> 4) << 1;
  const int mhi  = (lane >> 4) << 3;      // C/D: M offset 0 or 8

  // ---- Phase 2: step-1 complex matmul (DFT-128 along l2) + twiddle ------
  const float W128 = PI2F / 128.0f;
  const float WTW  = PI2F / (float)LLEN;
  for (int i = 0; i < 4; ++i) {
    const int tile = wave * 4 + i;        // 32 tiles: 4 (l1) x 8 (t2)
    const int m0 = (tile >> 3) * 16;      // l1 tile base
    const int n0 = (tile & 7) * 16;       // t2 tile base
    const int row = m0 + frow;            // A row (l1)
    const int col = n0 + frow;            // B col (t2)
    v8f gr = {0.f, 0.f, 0.f, 0.f, 0.f, 0.f, 0.f, 0.f};
    v8f gi = {0.f, 0.f, 0.f, 0.f, 0.f, 0.f, 0.f, 0.f};
    for (int k0 = 0; k0 < 128; k0 += 4) {
      const int kb = k0 + koff;
      v2f ar, ai, an;
      ar.x = sSr[row * N2 + kb]; ar.y = sSr[row * N2 + kb + 1];
      ai.x = sSi[row * N2 + kb]; ai.y = sSi[row * N2 + kb + 1];
      an = -ai;                            // f32 WMMA NEG is C-only -> negate A
      float s0, c0, s1, c1;                // exact reduction: period 128
      __sincosf(W128 * (float)((kb * col) & 127), &s0, &c0);
      __sincosf(W128 * (float)(((kb + 1) * col) & 127), &s1, &c1);
      v2f bc; bc.x = c0; bc.y = c1;
      v2f bs; bs.x = s0; bs.y = s1;
      gr = wmma4(ar, bc, gr);              // Gr += Sr*cos
      gr = wmma4(an, bs, gr);              // Gr -= Si*sin
      gi = wmma4(ar, bs, gi);              // Gi += Sr*sin
      gi = wmma4(ai, bc, gi);              // Gi += Si*cos
    }
    // twiddle e^{+2pi i l1 t2 / L} and park in LDS
    for (int r = 0; r < 8; ++r) {
      const int l1 = m0 + r + mhi;
      const int t2 = n0 + (lane & 15);
      float s, c;
      __sincosf(WTW * (float)((l1 * t2) & (LLEN - 1)), &s, &c);
      const float xr = gr[r], xi = gi[r];
      sGr[l1 * N2 + t2] = xr * c - xi * s;
      sGi[l1 * N2 + t2] = xr * s + xi * c;
    }
  }
  __syncthreads();

  // ---- Phase 3: step-2 matmul (DFT-64 along l1), real part only ---------
  const float W64 = PI2F / 64.0f;
  for (int i = 0; i < 4; ++i) {
    const int tile = wave * 4 + i;        // 32 tiles: 4 (t1) x 8 (t2)
    const int m0 = (tile >> 3) * 16;      // t1 tile base
    const int n0 = (tile & 7) * 16;       // t2 tile base
    const int row = m0 + frow;            // A row (t1)
    const int col = n0 + frow;            // B col (t2)
    v8f acc = {0.f, 0.f, 0.f, 0.f, 0.f, 0.f, 0.f, 0.f};
    for (int k0 = 0; k0 < 64; k0 += 4) {
      const int kb = k0 + koff;
      float s0, c0, s1, c1;                // exact reduction: period 64
      __sincosf(W64 * (float)((row * kb) & 63), &s0, &c0);
      __sincosf(W64 * (float)((row * (kb + 1)) & 63), &s1, &c1);
      v2f ac; ac.x = c0;  ac.y = c1;
      v2f as; as.x = -s0; as.y = -s1;
      v2f br_, bi_;
      br_.x = sGr[kb * N2 + col]; br_.y = sGr[(kb + 1) * N2 + col];
      bi_.x = sGi[kb * N2 + col]; bi_.y = sGi[(kb + 1) * N2 + col];
      acc = wmma4(ac, br_, acc);           // Kr += cos*Gr
      acc = wmma4(as, bi_, acc);           // Kr -= sin*Gi
    }
    for (int r = 0; r < 8; ++r) {
      const int t1 = m0 + r + mhi;
      const int t2 = n0 + (lane & 15);
      out[h * LLEN + t1 * N2 + t2] = acc[r];  // t = t2 + 128*t1
    }
  }
}

extern "C" void kernel_launch(void* const* d_in, const int* in_sizes, int n_in,
                              void* d_out, int out_size, void* d_ws, size_t ws_size,
                              hipStream_t stream) {
  (void)in_sizes; (void)n_in; (void)out_size; (void)d_ws; (void)ws_size;
  const float* w_ri   = (const float*)d_in[0];
  const float* p_ri   = (const float*)d_in[1];
  const float* q_ri   = (const float*)d_in[2];
  const float* B_ri   = (const float*)d_in[3];
  const float* C_ri   = (const float*)d_in[4];
  const float* log_dt = (const float*)d_in[5];
  // d_in[6] is L (=8192), compile-time constant here.
  float* out = (float*)d_out;
  s4_nplr_fused<<<dim3(HH), dim3(256), 0, stream>>>(
      w_ri, p_ri, q_ri, B_ri, C_ri, log_dt, out);
}